// MultiHeadSelfAttention_77283641524499
// MI455X (gfx1250) — compile-verified
//
#include <hip/hip_runtime.h>
#include <hip/hip_bf16.h>

typedef float v2f __attribute__((ext_vector_type(2)));
typedef float v4f __attribute__((ext_vector_type(4)));
typedef float v8f __attribute__((ext_vector_type(8)));

// ---------------------------------------------------------------------------
// Tensor Data Mover helper: 2-D tile DMA global -> LDS (D# packed per
// CDNA5 ISA; count=1, type=2, data_size=4B,
// no padding, no multicast, no iteration).
// ---------------------------------------------------------------------------
#if __has_builtin(__builtin_amdgcn_tensor_load_to_lds)
#define HAVE_TDM 1
typedef unsigned int u32x4 __attribute__((ext_vector_type(4)));
typedef int i32x4 __attribute__((ext_vector_type(4)));
typedef int i32x8 __attribute__((ext_vector_type(8)));

__device__ __forceinline__ void tdm_load_2d_f32(
    const float* gsrc, unsigned lds_off,
    unsigned tensor_w, unsigned tensor_h,
    unsigned tile_w, unsigned tile_h, unsigned row_stride)
{
    unsigned long long ga = (unsigned long long)gsrc;
    u32x4 g0;
    g0.x = 1u;                                        // count=1 (valid D#)
    g0.y = lds_off;                                   // lds_addr [63:32]
    g0.z = (unsigned)(ga & 0xFFFFFFFFu);              // global_addr lo
    g0.w = (unsigned)((ga >> 32) & 0x1FFFFFFu)        // global_addr hi (57b)
         | (2u << 30);                                // type=2 ("image")
    i32x8 g1;
    g1[0] = (int)(2u << 16);                          // data_size=2 -> 4 bytes
    g1[1] = (int)((tensor_w & 0xFFFFu) << 16);        // tensor_dim0[15:0]
    g1[2] = (int)(((tensor_w >> 16) & 0xFFFFu)        // tensor_dim0[31:16]
         | ((tensor_h & 0xFFFFu) << 16));             // tensor_dim1[15:0]
    g1[3] = (int)(((tensor_h >> 16) & 0xFFFFu)        // tensor_dim1[31:16]
         | ((tile_w & 0xFFFFu) << 16));               // tile_dim0
    g1[4] = (int)(tile_h & 0xFFFFu);                  // tile_dim1 (tile_dim2=0)
    g1[5] = (int)row_stride;                          // tensor_dim0_stride lo
    g1[6] = 0;                                        // stride hi / dim1_stride
    g1[7] = 0;
    i32x4 z4 = {0, 0, 0, 0};
#if __clang_major__ >= 23
    i32x8 z8 = {0, 0, 0, 0, 0, 0, 0, 0};
    __builtin_amdgcn_tensor_load_to_lds(g0, g1, z4, z4, z8, 0);
#else
    __builtin_amdgcn_tensor_load_to_lds(g0, g1, z4, z4, 0);
#endif
}

__device__ __forceinline__ void tdm_wait() {
#if __has_builtin(__builtin_amdgcn_s_wait_tensorcnt)
    __builtin_amdgcn_s_wait_tensorcnt(0);
#else
    asm volatile("s_wait_tensorcnt 0x0" ::: "memory");
#endif
}

__device__ __forceinline__ unsigned lds_offset(const void* p) {
    // generic LDS pointer: low 32 bits are the wave-relative LDS byte offset
    return (unsigned)(unsigned long long)p;
}
#endif

// ---------------------------------------------------------------------------
// fp32 GEMM:  C[M,N] = A[M,K] @ B[K,N] (+bias).  V_WMMA_F32_16X16X4_F32.
// 256 threads = 8 wave32; block tile 64x128, BK=16, double-buffered LDS.
// Wave (wid&3) -> rows, (wid>>2) -> 64-wide column strip (4 accumulators
// sharing one A fragment per k-step).  Tiles staged by the TDM when
// available (wave 0 issues two 2-D descriptors, s_wait_tensorcnt + barrier),
// else by register-prefetch double buffering.
// ---------------------------------------------------------------------------
#define BM 64
#define BN 128
#define BK 16

__global__ __launch_bounds__(256) void gemm_f32_wmma(
    const float* __restrict__ A, const float* __restrict__ B,
    float* __restrict__ C, int M, int N, int K, const float* __restrict__ bias)
{
    __shared__ float lsA[2][BM][BK];   // 2 x 4 KB
    __shared__ float lsB[2][BK][BN];   // 2 x 8 KB

    const int tid   = threadIdx.x;
    const int wid   = tid >> 5;
    const int lane  = tid & 31;
    const int lhalf = lane >> 4;
    const int l16   = lane & 15;

    const int bm = blockIdx.y * BM;
    const int bn = blockIdx.x * BN;
    const int wm = (wid & 3) * 16;
    const int wn = (wid >> 2) * 64;

    v8f acc[4];
#pragma unroll
    for (int nt = 0; nt < 4; ++nt) acc[nt] = (v8f){};

    const int T = K / BK;

#if defined(HAVE_TDM)
    if (wid == 0) {
        tdm_load_2d_f32(A + (size_t)bm * K, lds_offset(&lsA[0][0][0]),
                        K, M, BK, BM, K);
        tdm_load_2d_f32(B + bn, lds_offset(&lsB[0][0][0]),
                        N, K, BN, BK, N);
        tdm_wait();
    }
    __syncthreads();
#else
    const int arow = tid >> 2;          // 0..63
    const int acol = (tid & 3) * 4;     // 0,4,8,12
    const int brow = tid >> 4;          // 0..15
    const int bcol = (tid & 15) * 8;    // 0..120 (two float4 each)
    *(v4f*)&lsA[0][arow][acol] = *(const v4f*)(A + (size_t)(bm + arow) * K + acol);
    *(v4f*)&lsB[0][brow][bcol] = *(const v4f*)(B + (size_t)brow * N + bn + bcol);
    *(v4f*)&lsB[0][brow][bcol + 4] = *(const v4f*)(B + (size_t)brow * N + bn + bcol + 4);
    __syncthreads();
#endif

    for (int t = 0; t < T; ++t) {
        const int cur = t & 1;
        const int nxt = cur ^ 1;

#if defined(HAVE_TDM)
        if (t + 1 < T && wid == 0) {
            const int k1 = (t + 1) * BK;
            tdm_load_2d_f32(A + (size_t)bm * K + k1, lds_offset(&lsA[nxt][0][0]),
                            K, M, BK, BM, K);
            tdm_load_2d_f32(B + (size_t)k1 * N + bn, lds_offset(&lsB[nxt][0][0]),
                            N, K, BN, BK, N);
        }
#else
        v4f pa, pb0, pb1;
        if (t + 1 < T) {
            const int k1 = (t + 1) * BK;
            pa  = *(const v4f*)(A + (size_t)(bm + arow) * K + k1 + acol);
            pb0 = *(const v4f*)(B + (size_t)(k1 + brow) * N + bn + bcol);
            pb1 = *(const v4f*)(B + (size_t)(k1 + brow) * N + bn + bcol + 4);
        }
#endif

#pragma unroll
        for (int s = 0; s < BK / 4; ++s) {
            // A frag (16x4): (row = wm+l16, k = 4s + 2*lhalf .. +1)
            v2f af = *(const v2f*)&lsA[cur][wm + l16][4 * s + 2 * lhalf];
#pragma unroll
            for (int nt = 0; nt < 4; ++nt) {
                v2f bf;
                bf.x = lsB[cur][4 * s + 2 * lhalf][wn + nt * 16 + l16];
                bf.y = lsB[cur][4 * s + 2 * lhalf + 1][wn + nt * 16 + l16];
                acc[nt] = __builtin_amdgcn_wmma_f32_16x16x4_f32(
                    false, af, false, bf, (short)0, acc[nt], false, false);
            }
        }

#if defined(HAVE_TDM)
        if (t + 1 < T && wid == 0) tdm_wait();
#else
        if (t + 1 < T) {
            *(v4f*)&lsA[nxt][arow][acol]     = pa;
            *(v4f*)&lsB[nxt][brow][bcol]     = pb0;
            *(v4f*)&lsB[nxt][brow][bcol + 4] = pb1;
        }
#endif
        __syncthreads();
    }

    // C/D layout: VGPR j -> row j + 8*lhalf, col l16
#pragma unroll
    for (int j = 0; j < 8; ++j) {
        const int row = bm + wm + j + 8 * lhalf;
#pragma unroll
        for (int nt = 0; nt < 4; ++nt) {
            const int col = bn + wn + nt * 16 + l16;
            float v = acc[nt][j];
            if (bias) v += bias[col];
            C[(size_t)row * N + col] = v;
        }
    }
}

// ---------------------------------------------------------------------------
// Causal flash attention, fp32 WMMA, one wave32 per (b, h, 16-row q tile).
// Online softmax; P bounced through LDS to convert C-layout -> A-layout.
// ---------------------------------------------------------------------------
__global__ __launch_bounds__(32) void attn_f32_wmma(
    const float* __restrict__ Q, const float* __restrict__ K,
    const float* __restrict__ V, float* __restrict__ Z,
    int S, int D, int H)
{
    const int hd    = D / H;            // 64
    const int qt    = blockIdx.x;
    const int h     = blockIdx.y;
    const int b     = blockIdx.z;
    const int lane  = threadIdx.x;
    const int lhalf = lane >> 4;
    const int l16   = lane & 15;

    const size_t base = (size_t)b * S * D + (size_t)h * hd;
    const float* Qb = Q + base;
    const float* Kb = K + base;
    const float* Vb = V + base;
    float*       Zb = Z + base;

    const int q0 = qt * 16;
    const float scale = 0.125f;         // 1/sqrt(64)
    const float NEG_INF = -__builtin_inff();

    __shared__ float lsP[16][16];

    v2f qa[16];
#pragma unroll
    for (int s = 0; s < 16; ++s)
        qa[s] = *(const v2f*)(Qb + (size_t)(q0 + l16) * D + 4 * s + 2 * lhalf);

    v8f acc[4];
#pragma unroll
    for (int nt = 0; nt < 4; ++nt) acc[nt] = (v8f){};
    float mrow[8], lrow[8];
#pragma unroll
    for (int j = 0; j < 8; ++j) { mrow[j] = NEG_INF; lrow[j] = 0.0f; }

    for (int kt = 0; kt <= qt; ++kt) {
        const int kr = kt * 16;

        v8f sfr = (v8f){};
#pragma unroll
        for (int s = 0; s < 16; ++s) {
            v2f bf = *(const v2f*)(Kb + (size_t)(kr + l16) * D + 4 * s + 2 * lhalf);
            sfr = __builtin_amdgcn_wmma_f32_16x16x4_f32(
                false, qa[s], false, bf, (short)0, sfr, false, false);
        }

        float p[8];
#pragma unroll
        for (int j = 0; j < 8; ++j) {
            float v = sfr[j] * scale;
            const int m = j + 8 * lhalf;
            if (kt == qt && l16 > m) v = NEG_INF;

            float mx = v;
            mx = fmaxf(mx, __shfl_xor(mx, 8, 32));
            mx = fmaxf(mx, __shfl_xor(mx, 4, 32));
            mx = fmaxf(mx, __shfl_xor(mx, 2, 32));
            mx = fmaxf(mx, __shfl_xor(mx, 1, 32));

            const float mnew = fmaxf(mrow[j], mx);
            const float corr = __expf(mrow[j] - mnew);
            mrow[j] = mnew;

            const float pv = __expf(v - mnew);
            float sm = pv;
            sm += __shfl_xor(sm, 8, 32);
            sm += __shfl_xor(sm, 4, 32);
            sm += __shfl_xor(sm, 2, 32);
            sm += __shfl_xor(sm, 1, 32);

            lrow[j] = lrow[j] * corr + sm;
            p[j] = pv;
            acc[0][j] *= corr; acc[1][j] *= corr;
            acc[2][j] *= corr; acc[3][j] *= corr;
        }

        __syncthreads();
#pragma unroll
        for (int j = 0; j < 8; ++j)
            lsP[j + 8 * lhalf][l16] = p[j];
        __syncthreads();
        v2f pa[4];
#pragma unroll
        for (int s = 0; s < 4; ++s)
            pa[s] = *(const v2f*)&lsP[l16][4 * s + 2 * lhalf];

#pragma unroll
        for (int nt = 0; nt < 4; ++nt) {
#pragma unroll
            for (int s = 0; s < 4; ++s) {
                const int krow = kr + 4 * s + 2 * lhalf;
                v2f vb;
                vb.x = Vb[(size_t)krow * D + nt * 16 + l16];
                vb.y = Vb[(size_t)(krow + 1) * D + nt * 16 + l16];
                acc[nt] = __builtin_amdgcn_wmma_f32_16x16x4_f32(
                    false, pa[s], false, vb, (short)0, acc[nt], false, false);
            }
        }
        __syncthreads();
    }

#pragma unroll
    for (int j = 0; j < 8; ++j) {
        const float inv = 1.0f / lrow[j];
        const int row = q0 + j + 8 * lhalf;
#pragma unroll
        for (int nt = 0; nt < 4; ++nt)
            Zb[(size_t)row * D + nt * 16 + l16] = acc[nt][j] * inv;
    }
}

// ---------------------------------------------------------------------------
// Launch: QKV projections -> flash attention -> output projection (+bias)
// ---------------------------------------------------------------------------
extern "C" void kernel_launch(void* const* d_in, const int* in_sizes, int n_in,
                              void* d_out, int out_size, void* d_ws, size_t ws_size,
                              hipStream_t stream) {
    const float* X  = (const float*)d_in[0];
    const float* Wq = (const float*)d_in[1];
    const float* Wk = (const float*)d_in[2];
    const float* Wv = (const float*)d_in[3];
    const float* Wo = (const float*)d_in[4];
    const float* bo = (const float*)d_in[5];

    const int B = 2, S = 2048, D = 1024, H = 16;
    const int M = B * S;                     // 4096

    float* Qbuf = (float*)d_ws;
    float* Kbuf = Qbuf + (size_t)M * D;
    float* Vbuf = Kbuf + (size_t)M * D;
    float* Zbuf = Vbuf + (size_t)M * D;      // 4 x 16 MB = 64 MB total

    dim3 gg(D / BN, M / BM);                 // (8, 64)
    gemm_f32_wmma<<<gg, 256, 0, stream>>>(X, Wq, Qbuf, M, D, D, nullptr);
    gemm_f32_wmma<<<gg, 256, 0, stream>>>(X, Wk, Kbuf, M, D, D, nullptr);
    gemm_f32_wmma<<<gg, 256, 0, stream>>>(X, Wv, Vbuf, M, D, D, nullptr);

    attn_f32_wmma<<<dim3(S / 16, H, B), 32, 0, stream>>>(
        Qbuf, Kbuf, Vbuf, Zbuf, S, D, H);

    gemm_f32_wmma<<<gg, 256, 0, stream>>>(Zbuf, Wo, (float*)d_out, M, D, D, bo);
}